// MRConv1d_74002286510469
// MI455X (gfx1250) — compile-verified
//
#include <hip/hip_runtime.h>

typedef __attribute__((ext_vector_type(2))) float v2f;
typedef __attribute__((ext_vector_type(8))) float v8f;

#define Bn    4
#define Nn    10000
#define Cn    128
#define Kn    16
#define OUTn  128
#define TOTAL (Bn * Nn)          // 40000 rows
#define BNK   (Bn * Nn * Kn)     // offset of edge_index[1] plane
#define TWO_C (2 * Cn)           // 256
#define FSTRIDE 258              // 256 + 2 pad floats: even, bank-conflict-free
#define TILE  128                // rows per block

__global__ __launch_bounds__(256) void mrconv_wmma_kernel(
    const float* __restrict__ x,      // [B, N, C]
    const int*   __restrict__ edge,   // [2, B, N, K]
    const float* __restrict__ W,      // [OUT, 2C]
    const float* __restrict__ bias,   // [OUT]
    float*       __restrict__ out)    // [B, N, OUT]
{
    __shared__ float feat[TILE * FSTRIDE];   // 132096 B
    __shared__ int   sidx[TILE * 2 * Kn];    //  16384 B

    const int base = blockIdx.x * TILE;
    const int tid  = threadIdx.x;

    // ---- stage all neighbor indices for this tile into LDS (coalesced) ----
    for (int i = tid; i < TILE * 2 * Kn; i += 256) {
        const int nl = i >> 5;        // node-local 0..127
        const int kk = i & 31;        // 0..15 -> edge[0], 16..31 -> edge[1]
        const int g  = base + nl;
        int v = 0;
        if (g < TOTAL)
            v = (kk < Kn) ? edge[g * Kn + kk] : edge[BNK + g * Kn + (kk - Kn)];
        sidx[i] = v;
    }
    __syncthreads();

    // ---- phase 1: build interleaved [x_c, rel_c] feature rows in LDS ----
    // wave = one node, 32 contiguous channels -> 128B coalesced gathers (L2 hits)
    const int c    = tid & (Cn - 1);  // channel 0..127
    const int slot = tid >> 7;        // which of 2 nodes in flight
    for (int iter = 0; iter < TILE / 2; ++iter) {
        const int nl = iter * 2 + slot;
        const int g  = base + nl;
        float xv = 0.f, rel = 0.f;
        if (g < TOTAL) {
            const int bb = g / Nn;
            const float* xb = x + (size_t)bb * Nn * Cn + c;
            xv  = x[(size_t)g * Cn + c];
            rel = -3.402823466e38f;
            const int* id = sidx + nl * 2 * Kn;
#pragma unroll
            for (int k = 0; k < Kn; ++k) {
                // rel = max_k( x_j - x_i ), j = edge[0], i = edge[1]
                rel = fmaxf(rel, xb[(size_t)id[k] * Cn] - xb[(size_t)id[Kn + k] * Cn]);
            }
        }
        v2f f; f.x = xv; f.y = rel;                   // interleave [x_c, rel_c]
        *(v2f*)&feat[nl * FSTRIDE + 2 * c] = f;       // ds_store_b64, no conflicts
    }
    __syncthreads();

    // ---- phase 2: fp32 WMMA GEMM  out[128 x 128] = feat[128 x 256] @ W^T ----
    // wave w owns output columns [16w, 16w+16) across all 8 row tiles
    const int lane = tid & 31;
    const int wave = tid >> 5;
    const int o0   = wave * 16;
    const int ocol = lane & 15;       // column within 16-wide N tile
    const int half = lane >> 4;       // selects K pair (A/B) / row half (C/D)

    v8f acc[8] = {};                  // 8 M-tiles x (16x16 f32 / 32 lanes)

    const float* wrow = W + (size_t)(o0 + ocol) * TWO_C;   // B[k][o] = W[o][k]
    for (int ks = 0; ks < TWO_C / 4; ++ks) {               // 64 K-steps of 4
        const int kb = ks * 4 + half * 2;
        const v2f bfrag = *(const v2f*)(wrow + kb);        // reused by 8 WMMAs
#pragma unroll
        for (int mt = 0; mt < 8; ++mt) {
            const v2f afrag = *(const v2f*)&feat[(mt * 16 + ocol) * FSTRIDE + kb];
            acc[mt] = __builtin_amdgcn_wmma_f32_16x16x4_f32(
                false, afrag, false, bfrag, (short)0, acc[mt], false, false);
        }
    }

    // ---- epilogue: bias + ReLU, guarded stores (tail tile) ----
    const float bv = bias[o0 + ocol];
#pragma unroll
    for (int mt = 0; mt < 8; ++mt) {
#pragma unroll
        for (int v = 0; v < 8; ++v) {
            const int row = base + mt * 16 + v + half * 8;  // C/D VGPR layout
            if (row < TOTAL) {
                float r = acc[mt][v] + bv;
                out[(size_t)row * OUTn + o0 + ocol] = r > 0.f ? r : 0.f;
            }
        }
    }
}

extern "C" void kernel_launch(void* const* d_in, const int* in_sizes, int n_in,
                              void* d_out, int out_size, void* d_ws, size_t ws_size,
                              hipStream_t stream) {
    const float* x    = (const float*)d_in[0];
    const int*   edge = (const int*)  d_in[1];
    const float* W    = (const float*)d_in[2];
    const float* b    = (const float*)d_in[3];
    float*       out  = (float*)d_out;

    const int nblocks = (TOTAL + TILE - 1) / TILE;   // 313
    mrconv_wmma_kernel<<<nblocks, 256, 0, stream>>>(x, edge, W, b, out);
}